// KerasArima_57681410786206
// MI455X (gfx1250) — compile-verified
//
#include <hip/hip_runtime.h>

typedef __attribute__((ext_vector_type(2))) float v2f;
typedef __attribute__((ext_vector_type(8))) float v8f;

#define T_DIM  2048
#define HW     256
#define LAST_T 2047
#define CHUNK_STRIDE (16 * HW)   // floats per 16-timestep chunk

#define WMMA_F32_4(Amat, Bmat, Cmat)                                          \
  __builtin_amdgcn_wmma_f32_16x16x4_f32(false, (Amat), false, (Bmat),         \
                                        (short)0, (Cmat), false, false)

struct Tile { v2f b0, b1, b2, b3; };

// Swap the two 16-lane halves on the VALU pipe (identity selects).
// FI=1: the compiler may sink this into an EXEC-predicated block where the
// source half is inactive; fetch-inactive makes those reads defined.
__device__ __forceinline__ float permx16(float v) {
  int iv = __builtin_bit_cast(int, v);
  int sw = __builtin_amdgcn_permlanex16(iv, iv, 0x76543210, 0xfedcba98,
                                        true, false);
  return __builtin_bit_cast(float, sw);
}

// Load one 16(t) x 16(seq) tile of x in the 4x16 B-matrix slice layout:
// slice k, VGPR v, lane l  ->  x[t0 + 4k + v + 2*(l>=16), hw0 + (l&15)]
// p = xb + (t0 + 2*hi)*HW + col. Non-temporal: x is streamed exactly once.
__device__ __forceinline__ Tile load_tile(const float* __restrict__ p) {
  Tile r;
  r.b0.x = __builtin_nontemporal_load(p + 0 * HW);
  r.b0.y = __builtin_nontemporal_load(p + 1 * HW);
  r.b1.x = __builtin_nontemporal_load(p + 4 * HW);
  r.b1.y = __builtin_nontemporal_load(p + 5 * HW);
  r.b2.x = __builtin_nontemporal_load(p + 8 * HW);
  r.b2.y = __builtin_nontemporal_load(p + 9 * HW);
  r.b3.x = __builtin_nontemporal_load(p + 12 * HW);
  r.b3.y = __builtin_nontemporal_load(p + 13 * HW);
  return r;
}

// Tail chunk (t0 = 2034): clamp t to LAST_T. Clamped lanes only feed L'
// entries that are exactly 0 (j > i for every stored row).
__device__ __forceinline__ Tile load_tile_tail(const float* __restrict__ xb,
                                               int hi, int col) {
  const int tb = 2034 + 2 * hi;
  auto ld = [&](int tl) {
    int t = tb + tl; t = (t > LAST_T) ? LAST_T : t;
    return __builtin_nontemporal_load(xb + (size_t)t * HW + col);
  };
  Tile r;
  r.b0.x = ld(0);  r.b0.y = ld(1);
  r.b1.x = ld(4);  r.b1.y = ld(5);
  r.b2.x = ld(8);  r.b2.y = ld(9);
  r.b3.x = ld(12); r.b3.y = ld(13);
  return r;
}

__global__ __launch_bounds__(32)
void arima_wmma_scan(const float* __restrict__ x,
                     const float* __restrict__ pphi,
                     const float* __restrict__ pth1,
                     const float* __restrict__ pth2,
                     const float* __restrict__ pe0,
                     float* __restrict__ y) {
  __shared__ float h_ext[32];   // h_ext[16+d] = h(d), zero for d < 0
  __shared__ float alp[16];     // alpha_i : coefficient of carry y_{t0-1}
  __shared__ float bet[16];     // beta_i  : coefficient of carry y_{t0-2}

  const float phi = pphi[0], th1 = pth1[0], th2 = pth2[0], e0 = pe0[0];
  const float ca = 1.0f + phi - th1;   // g_t = ca*x_t + cb*x_{t-1}
  const float cb = -phi - th2;

  if (threadIdx.x == 0) {
    for (int d = 0; d < 16; ++d) h_ext[d] = 0.0f;
    h_ext[16] = 1.0f;                       // h(0)
    float hp = 1.0f, hpp = 0.0f;
    for (int d = 1; d < 16; ++d) { float hn = th1 * hp + th2 * hpp; h_ext[16 + d] = hn; hpp = hp; hp = hn; }
    float ap = 1.0f, app = 0.0f;            // alpha_{-1}=1, alpha_{-2}=0
    for (int t = 0; t < 16; ++t) { float an = th1 * ap + th2 * app; alp[t] = an; app = ap; ap = an; }
    float bp = 0.0f, bpp = 1.0f;            // beta_{-1}=0,  beta_{-2}=1
    for (int t = 0; t < 16; ++t) { float bn = th1 * bp + th2 * bpp; bet[t] = bn; bpp = bp; bp = bn; }
  }
  __syncthreads();

  const int lane  = threadIdx.x & 31;
  const int m     = lane & 15;                 // A row / D column index
  const int hi    = lane >> 4;                 // lane half
  const int gwave = blockIdx.x;                // one wave per workgroup
  const int batch = gwave >> 4;                // 16 waves per batch
  const int hw0   = (gwave & 15) << 4;         // 16 sequences per wave
  const int col   = hw0 + m;

  const float* xb = x + (size_t)batch * T_DIM * HW;
  float*       yb = y + (size_t)batch * T_DIM * HW;

  // Constant A-matrix slices of L' in WMMA 16x4 layout:
  // slice k, VGPR v, lane l -> L'[l&15, 4k + v + 2*(l>=16)]
  v2f A0, A1, A2, A3, Ab;
  {
    const int j0 = 2 * hi;
    auto Lp = [&](int i, int j) {
      return ca * h_ext[16 + i - j] + cb * h_ext[15 + i - j];
    };
    A0.x = Lp(m, j0 + 0);  A0.y = Lp(m, j0 + 1);
    A1.x = Lp(m, j0 + 4);  A1.y = Lp(m, j0 + 5);
    A2.x = Lp(m, j0 + 8);  A2.y = Lp(m, j0 + 9);
    A3.x = Lp(m, j0 + 12); A3.y = Lp(m, j0 + 13);
    // Boundary 16x4: columns = [alpha, beta, cb*h, 0]
    Ab.x = hi ? (cb * h_ext[16 + m]) : alp[m];
    Ab.y = hi ? 0.0f : bet[m];
  }

  // y0, y1 special cases (U == 0). Lanes l and l+16 read the same column,
  // so the initial boundary B operand is built with no shuffles at all.
  const float x0v = xb[col];
  const float x1v = xb[HW + col];
  const float y0v = x0v - th1 * e0;
  const float y1v = x1v + phi * (x1v - x0v) - th1 * (x1v - y0v) - th2 * e0;
  if (hi == 0) { yb[col] = y0v; yb[HW + col] = y1v; }

  // Loop-carried boundary B 4x16, rows = [y_{t0-1}; y_{t0-2}; x_{t0-1}; 0]:
  //   Bb.x : lo lanes = y_{t0-1}[col], hi lanes = x_{t0-1}[col]
  //   Bb.y : lo lanes = y_{t0-2}[col], hi lanes = 0
  v2f Bb;
  Bb.x = hi ? x1v : y1v;
  Bb.y = hi ? 0.0f : y0v;

  const float* px = xb + (size_t)(2 + 2 * hi) * HW + col;   // chunk-0 load base
  float*       qy = yb + (size_t)(2 + 8 * hi) * HW + col;   // chunk-0 store base

  Tile cur = load_tile(px);

  // Chunks 0..125: the next chunk is always a full in-bounds tile.
  for (int c = 0; c < 126; ++c) {
    Tile nxt = load_tile(px + (size_t)(c + 1) * CHUNK_STRIDE);
    __builtin_prefetch(px + (size_t)(c + 2) * CHUNK_STRIDE, 0, 1);

    // Data contributions first: they depend only on loads, so the serial
    // carry->carry path is exactly one WMMA (the boundary one, last).
    v8f C = {};
    C = WMMA_F32_4(A0, cur.b0, C);
    C = WMMA_F32_4(A1, cur.b1, C);
    C = WMMA_F32_4(A2, cur.b2, C);
    C = WMMA_F32_4(A3, cur.b3, C);
    C = WMMA_F32_4(Ab, Bb, C);

    // Next boundary operand: hi lanes already hold x[t0+15] (cur.b3.y);
    // lo lanes fetch y rows 15/14 from the hi half of D (VGPR 7/6).
    Bb.x = hi ? cur.b3.y : permx16(C[7]);
    Bb.y = hi ? 0.0f     : permx16(C[6]);

    float* q = qy + (size_t)c * CHUNK_STRIDE;
#pragma unroll
    for (int r = 0; r < 8; ++r) __builtin_nontemporal_store(C[r], q + r * HW);
    cur = nxt;
  }

  // Chunk 126 (full), overlapping the clamped loads of tail chunk 127.
  {
    Tile nxt = load_tile_tail(xb, hi, col);

    v8f C = {};
    C = WMMA_F32_4(A0, cur.b0, C);
    C = WMMA_F32_4(A1, cur.b1, C);
    C = WMMA_F32_4(A2, cur.b2, C);
    C = WMMA_F32_4(A3, cur.b3, C);
    C = WMMA_F32_4(Ab, Bb, C);

    Bb.x = hi ? cur.b3.y : permx16(C[7]);
    Bb.y = hi ? 0.0f     : permx16(C[6]);

    float* q = qy + (size_t)126 * CHUNK_STRIDE;
#pragma unroll
    for (int r = 0; r < 8; ++r) __builtin_nontemporal_store(C[r], q + r * HW);
    cur = nxt;
  }

  // Chunk 127 (t0 = 2034): only rows with t <= 2047 are stored.
  {
    v8f C = {};
    C = WMMA_F32_4(A0, cur.b0, C);
    C = WMMA_F32_4(A1, cur.b1, C);
    C = WMMA_F32_4(A2, cur.b2, C);
    C = WMMA_F32_4(A3, cur.b3, C);
    C = WMMA_F32_4(Ab, Bb, C);

    float* q = qy + (size_t)127 * CHUNK_STRIDE;
#pragma unroll
    for (int r = 0; r < 8; ++r) {
      if (8 * hi + r <= 13)   // t = 2034 + 8*hi + r <= LAST_T
        __builtin_nontemporal_store(C[r], q + r * HW);
    }
  }
}

extern "C" void kernel_launch(void* const* d_in, const int* in_sizes, int n_in,
                              void* d_out, int out_size, void* d_ws, size_t ws_size,
                              hipStream_t stream) {
  const float* x   = (const float*)d_in[0];
  const float* phi = (const float*)d_in[1];
  const float* th1 = (const float*)d_in[2];
  const float* th2 = (const float*)d_in[3];
  const float* e0  = (const float*)d_in[4];
  float* y = (float*)d_out;

  // 16384 sequences / 16 per wave = 1024 waves, one wave per workgroup
  // so the scheduler can spread them across every WGP.
  dim3 grid(1024), block(32);
  hipLaunchKernelGGL(arima_wmma_scan, grid, block, 0, stream,
                     x, phi, th1, th2, e0, y);
}